// MultipleKernelMaximumMeanDiscrepancy_54331336294978
// MI455X (gfx1250) — compile-verified
//
#include <hip/hip_runtime.h>
#include <math.h>

#define Bn 2048
#define Dn 1024
#define Nn 4096   // 2*Bn

typedef __attribute__((ext_vector_type(2))) float v2f;
typedef __attribute__((ext_vector_type(8))) float v8f;

// ws layout (floats):
//   [0, 1024)        : colsum  (sum over rows of each column of concat(z_s,z_t))
//   [1024, 5120)     : sq      (per-row squared norms, concat order)
//   [5120]           : mean_d2
//   [5121]           : loss accumulator

// ---------------- init: zero colsum + loss accumulator ----------------
__global__ void init_ws_kernel(float* ws) {
    int t = blockIdx.x * blockDim.x + threadIdx.x;
    if (t < Dn) ws[t] = 0.0f;
    if (t == 0) ws[5121] = 0.0f;
}

// ---------------- per-row squared norms: one wave per row ----------------
__global__ void row_sq_kernel(const float* __restrict__ zs,
                              const float* __restrict__ zt,
                              float* __restrict__ sq) {
    int wave = (blockIdx.x * blockDim.x + threadIdx.x) >> 5;
    int lane = threadIdx.x & 31;
    if (wave >= Nn) return;
    const float* row = (wave < Bn) ? (zs + (size_t)wave * Dn)
                                   : (zt + (size_t)(wave - Bn) * Dn);
    float s = 0.0f;
#pragma unroll
    for (int j = 0; j < Dn / (32 * 4); ++j) {           // 8 x float4 per lane
        float4 v = ((const float4*)row)[lane + 32 * j]; // wave reads 512B contiguous
        s += v.x * v.x + v.y * v.y + v.z * v.z + v.w * v.w;
    }
#pragma unroll
    for (int off = 16; off > 0; off >>= 1) s += __shfl_xor(s, off);
    if (lane == 0) sq[wave] = s;
}

// ---------------- column sums: block = 256-row chunk, thread = 4 columns ----------------
__global__ void col_sum_kernel(const float* __restrict__ zs,
                               const float* __restrict__ zt,
                               float* __restrict__ colsum) {
    int c4   = threadIdx.x;            // columns 4*c4 .. 4*c4+3
    int row0 = blockIdx.x * 256;       // 16 blocks cover 4096 rows; 2048%256==0
    const float* basep = (row0 < Bn) ? (zs + (size_t)row0 * Dn)
                                     : (zt + (size_t)(row0 - Bn) * Dn);
    float4 acc = make_float4(0.f, 0.f, 0.f, 0.f);
    for (int r = 0; r < 256; ++r) {
        float4 v = ((const float4*)(basep + (size_t)r * Dn))[c4];
        acc.x += v.x; acc.y += v.y; acc.z += v.z; acc.w += v.w;
    }
    atomicAdd(&colsum[4 * c4 + 0], acc.x);
    atomicAdd(&colsum[4 * c4 + 1], acc.y);
    atomicAdd(&colsum[4 * c4 + 2], acc.z);
    atomicAdd(&colsum[4 * c4 + 3], acc.w);
}

// ---------------- mean_d2 = (2/N)*sum(sq) - (2/N^2)*||colsum||^2 ----------------
__global__ void finalize_stats_kernel(const float* __restrict__ colsum,
                                      const float* __restrict__ sq,
                                      float* __restrict__ mean_d2_out) {
    __shared__ float red[256];
    int t = threadIdx.x;
    float a = 0.0f;
    for (int i = t; i < Nn; i += 256) a += sq[i];
    float b = 0.0f;
    for (int i = t; i < Dn; i += 256) { float v = colsum[i]; b += v * v; }
    float c = a * (2.0f / (float)Nn) - b * (2.0f / ((float)Nn * (float)Nn));
    red[t] = c;
    __syncthreads();
#pragma unroll
    for (int s = 128; s > 0; s >>= 1) {
        if (t < s) red[t] += red[t + s];
        __syncthreads();
    }
    if (t == 0) mean_d2_out[0] = red[0];
}

// ---------------- sparse pair dots via V_WMMA_F32_16X16X4_F32 ----------------
// 512 waves; wave handles 16 pairs of one category:
//   q=0: +k(zs[i], zs[i+1])   q=1: +k(zt[i], zt[i+1])
//   q=2: -k(zs[i], zt[i+1])   q=3: -k(zs[i+1], zt[i])
// D = A * B^T accumulated over K=1024; diagonal D[m][m] = dot(left_m, right_m).
__global__ void pair_wmma_kernel(const float* __restrict__ zs,
                                 const float* __restrict__ zt,
                                 const float* __restrict__ sq,
                                 const float* __restrict__ mean_d2_p,
                                 float* __restrict__ loss_acc) {
    int wave = (blockIdx.x * blockDim.x + threadIdx.x) >> 5;  // 0..511
    int lane = threadIdx.x & 31;
    int q    = wave >> 7;        // category 0..3 (uniform per wave)
    int tloc = wave & 127;       // tile within category
    int m    = lane & 15;        // pair slot within tile
    int i    = tloc * 16 + m;    // pair index 0..2047
    int ip1  = (i + 1) & (Bn - 1);

    const float* lptr; const float* rptr; int gli, gri; float sign;
    switch (q) {
    case 0:  lptr = zs + (size_t)i   * Dn; rptr = zs + (size_t)ip1 * Dn;
             gli = i;        gri = ip1;        sign =  1.0f; break;
    case 1:  lptr = zt + (size_t)i   * Dn; rptr = zt + (size_t)ip1 * Dn;
             gli = Bn + i;   gri = Bn + ip1;   sign =  1.0f; break;
    case 2:  lptr = zs + (size_t)i   * Dn; rptr = zt + (size_t)ip1 * Dn;
             gli = i;        gri = Bn + ip1;   sign = -1.0f; break;
    default: lptr = zs + (size_t)ip1 * Dn; rptr = zt + (size_t)i   * Dn;
             gli = ip1;      gri = Bn + i;     sign = -1.0f; break;
    }

    // 16x4 f32 A layout: lanes 0-15 hold K=0..1, lanes 16-31 hold K=2..3 (same M rows).
    int koff = (lane >> 4) * 2;

    v8f acc0 = {};
    v8f acc1 = {};
    for (int k = 0; k < Dn; k += 8) {
        v2f a0 = *(const v2f*)(lptr + k + koff);
        v2f b0 = *(const v2f*)(rptr + k + koff);
        v2f a1 = *(const v2f*)(lptr + k + 4 + koff);
        v2f b1 = *(const v2f*)(rptr + k + 4 + koff);
        acc0 = __builtin_amdgcn_wmma_f32_16x16x4_f32(
            false, a0, false, b0, (short)0, acc0, false, false);
        acc1 = __builtin_amdgcn_wmma_f32_16x16x4_f32(
            false, a1, false, b1, (short)0, acc1, false, false);
    }
    v8f acc = acc0 + acc1;

    // Diagonal extraction per C/D layout:
    //   pair p<8  -> lane p      , vgpr p
    //   pair p>=8 -> lane p+16   , vgpr p-8
    float contrib = 0.0f;
    int vidx = (lane < 8) ? lane : ((lane >= 24) ? (lane - 24) : -1);
    if (vidx >= 0) {
        float dot = 0.0f;
#pragma unroll
        for (int r = 0; r < 8; ++r)
            if (vidx == r) dot = acc[r];
        // this lane's m == its diagonal pair slot (lane<8: m=lane; lane>=24: m=lane-16)
        float d2 = sq[gli] + sq[gri] - 2.0f * dot;
        d2 = fmaxf(d2, 0.0f);
        float md = mean_d2_p[0];
        float r1 = d2 / md;
        // alphas {0.5, 1, 2}: exp(-d2/(2*a*md)) = exp(-r1), exp(-r1/2), exp(-r1/4)
        contrib = sign * (expf(-r1) + expf(-0.5f * r1) + expf(-0.25f * r1));
    }
#pragma unroll
    for (int off = 16; off > 0; off >>= 1) contrib += __shfl_xor(contrib, off);
    if (lane == 0) atomicAdd(loss_acc, contrib);
}

// ---------------- final scale ----------------
__global__ void write_out_kernel(const float* __restrict__ loss_acc,
                                 float* __restrict__ out) {
    if (threadIdx.x == 0 && blockIdx.x == 0)
        out[0] = loss_acc[0] / (float)Bn;
}

extern "C" void kernel_launch(void* const* d_in, const int* in_sizes, int n_in,
                              void* d_out, int out_size, void* d_ws, size_t ws_size,
                              hipStream_t stream) {
    const float* zs = (const float*)d_in[0];
    const float* zt = (const float*)d_in[1];
    float* ws      = (float*)d_ws;
    float* colsum  = ws;            // 1024
    float* sq      = ws + 1024;     // 4096
    float* mean_d2 = ws + 5120;     // 1
    float* lossacc = ws + 5121;     // 1
    float* out     = (float*)d_out;

    init_ws_kernel      <<<4,   256, 0, stream>>>(ws);
    row_sq_kernel       <<<512, 256, 0, stream>>>(zs, zt, sq);       // 4096 waves
    col_sum_kernel      <<<16,  256, 0, stream>>>(zs, zt, colsum);
    finalize_stats_kernel<<<1,  256, 0, stream>>>(colsum, sq, mean_d2);
    pair_wmma_kernel    <<<64,  256, 0, stream>>>(zs, zt, sq, mean_d2, lossacc); // 512 waves
    write_out_kernel    <<<1,   32,  0, stream>>>(lossacc, out);
}